// MASNET2_55851754717595
// MI455X (gfx1250) — compile-verified
//
#include <hip/hip_runtime.h>
#include <hip/hip_bf16.h>

// MASNET attention resampler for MI455X (gfx1250, wave32).
// Stage 1: marginal maxes of structure_att (bandwidth bound, coalesced).
// Stage 2: per-batch 1D pipeline; 447-tap convs done as Toeplitz GEMM via
//          V_WMMA_F32_16X16X4_F32 (the only compute-dense stage).
// Stage 3: separable bilinear gather, staging input rows through LDS.

#define SAMP 224
#define INW  448
#define PADW 223
#define GLOBW 670      // SAMP + 2*PADW
#define KS   447
#define KBLK 116       // ceil(462/4) -> K loop of 116 blocks of 4
#define BATCH 64

typedef __attribute__((ext_vector_type(2))) float v2f;
typedef __attribute__((ext_vector_type(8))) float v8f;

// ---------------------------------------------------------------------------
// Kernel 1: per-batch column-max (axis=1) and row-max (axis=2) of att.
// 256 threads = 8 waves. Row maxes: wave handles rows i = wv, wv+8, ...
// (lane-strided coalesced reads + shfl_xor wave reduce). Column maxes:
// lane-local partials over 14 fixed columns, combined via LDS atomic max
// (values are uniform[0,1) >= 0, so int-compare max is order-preserving).
// ---------------------------------------------------------------------------
__global__ void masnet_reduce_kernel(const float* __restrict__ att,
                                     float* __restrict__ maps) {
  const int b = blockIdx.x;
  const int t = threadIdx.x;
  const int lane = t & 31;
  const int wv = t >> 5;

  __shared__ float s_col[INW];
  for (int j = t; j < INW; j += 256) s_col[j] = 0.0f;
  __syncthreads();

  float cmax[14];
#pragma unroll
  for (int k = 0; k < 14; ++k) cmax[k] = 0.0f;

  const float* base = att + (size_t)b * INW * INW;
  for (int i = wv; i < INW; i += 8) {
    const float* row = base + (size_t)i * INW;
    float rmax = 0.0f;
#pragma unroll
    for (int k = 0; k < 14; ++k) {
      float v = row[lane + 32 * k];
      rmax = fmaxf(rmax, v);
      cmax[k] = fmaxf(cmax[k], v);
    }
#pragma unroll
    for (int off = 16; off; off >>= 1)
      rmax = fmaxf(rmax, __shfl_xor(rmax, off, 32));
    if (lane == 0) maps[(size_t)b * 2 * INW + INW + i] = rmax;  // map_sy[i]
  }

#pragma unroll
  for (int k = 0; k < 14; ++k)
    atomicMax((int*)&s_col[lane + 32 * k], __float_as_int(cmax[k]));
  __syncthreads();

  for (int j = t; j < INW; j += 256)
    maps[(size_t)b * 2 * INW + j] = s_col[j];                   // map_sx[j]
}

// ---------------------------------------------------------------------------
// Kernel 2: per-batch 1D pipeline + Toeplitz-WMMA convolution.
// One block (256 threads = 8 waves) per batch.
//   out[ob+m] = sum_kp wext[kp-m] * sig[ob+kp],  kp in [0,464)
//   A[m][k]   = wext[kp-m]   (16x4 Toeplitz slice of the Gaussian filter)
//   B[k][n]   = sig_n[ob+kp] (n in {sx, P*sx, sy, P*sy}, cols 4..15 zero)
// accumulated over 116 K-blocks with V_WMMA_F32_16X16X4_F32.
// ---------------------------------------------------------------------------
__global__ void masnet_grid_kernel(const float* __restrict__ maps,
                                   const float* __restrict__ fw,
                                   float* __restrict__ coords) {
  __shared__ float s_map[2][INW];
  __shared__ float s_w[496];        // s_w[16+j] = fw[j] for j in [0,447), else 0
  __shared__ float s_sig[4][672];   // padded signals (zeros at 670..671)
  __shared__ float s_conv[4][SAMP];
  __shared__ float s_red[8];
  __shared__ float s_inv[2];

  const int b = blockIdx.x;
  const int t = threadIdx.x;
  const int lane = t & 31;
  const int wv = t >> 5;

  for (int j = t; j < INW; j += 256) {
    s_map[0][j] = maps[(size_t)b * 2 * INW + j];
    s_map[1][j] = maps[(size_t)b * 2 * INW + INW + j];
  }
  for (int j = t; j < 496; j += 256) {
    int k = j - 16;
    s_w[j] = (k >= 0 && k < KS) ? fw[k] : 0.0f;
  }
  __syncthreads();

  // normalization sums (interp is linear so scaling after interp is identical)
  for (int a = 0; a < 2; ++a) {
    float p = 0.0f;
    for (int j = t; j < INW; j += 256) p += s_map[a][j];
#pragma unroll
    for (int off = 16; off; off >>= 1) p += __shfl_xor(p, off, 32);
    if (lane == 0) s_red[wv] = p;
    __syncthreads();
    if (t == 0) {
      float s = 0.0f;
      for (int i = 0; i < 8; ++i) s += s_red[i];
      s_inv[a] = 1.0f / s;
    }
    __syncthreads();
  }

  // build interp(224) -> reflect-pad(670) signals, and P*signal
  for (int g = t; g < 672; g += 256) {
    for (int a = 0; a < 2; ++a) {
      float v = 0.0f, pv = 0.0f;
      if (g < GLOBW) {
        int p = g - PADW;                                  // [-223, 446]
        int r = (p < 0) ? -p : ((p > (SAMP - 1)) ? 2 * (SAMP - 1) - p : p);
        float pos = (float)r * (447.0f / 223.0f);          // interp position
        int i0 = (int)floorf(pos);
        int i1 = min(i0 + 1, INW - 1);
        float wf = pos - (float)i0;
        float m = s_map[a][i0] * (1.0f - wf) + s_map[a][i1] * wf;
        m *= s_inv[a];
        float P = (float)(g - PADW) / (float)(SAMP - 1);
        v = m;
        pv = P * m;
      }
      s_sig[2 * a + 0][g] = v;
      s_sig[2 * a + 1][g] = pv;
    }
  }
  __syncthreads();

  // Toeplitz WMMA conv: 14 output blocks of 16, distributed across 8 waves.
  // A (16x4 f32) lane layout per ISA 7.12.2: lanes 0-15 hold M=0-15 with
  // K=0 in v0 / K=1 in v1; lanes 16-31 hold K=2 / K=3.
  const int m = lane & 15;
  const int khalf = (lane >> 4) * 2;  // 0 or 2
  for (int ob16 = wv; ob16 < 14; ob16 += 8) {   // wave-uniform: EXEC all-ones
    const int ob = ob16 * 16;
    v8f acc = {};
    for (int tt = 0; tt < KBLK; ++tt) {
      const int kb = tt * 4 + khalf;
      v2f A, Bm;
      A.x = s_w[16 + kb - m];          // wext[kp - m], zero outside [0,447)
      A.y = s_w[16 + kb + 1 - m];
      const float* sp = (m < 4) ? s_sig[m] : s_sig[0];
      float b0 = sp[ob + kb];
      float b1 = sp[ob + kb + 1];
      Bm.x = (m < 4) ? b0 : 0.0f;
      Bm.y = (m < 4) ? b1 : 0.0f;
      acc = __builtin_amdgcn_wmma_f32_16x16x4_f32(
          /*neg_a=*/false, A, /*neg_b=*/false, Bm,
          /*c_mod=*/(short)0, acc, /*reuse_a=*/false, /*reuse_b=*/false);
    }
    // D layout: acc[v] holds M = v + 8*(lane>=16), N = lane&15
    if (m < 4) {
      const int row0 = (lane >> 4) * 8;
#pragma unroll
      for (int v = 0; v < 8; ++v) s_conv[m][ob + row0 + v] = acc[v];
    }
  }
  __syncthreads();

  for (int o = t; o < SAMP; o += 256) {
    float xf = s_conv[1][o] / s_conv[0][o];
    float yf = s_conv[3][o] / s_conv[2][o];
    float xg = fminf(fmaxf(xf * 2.0f - 1.0f, -1.0f), 1.0f);
    float yg = fminf(fmaxf(yf * 2.0f - 1.0f, -1.0f), 1.0f);
    coords[(size_t)b * 2 * SAMP + o] = xg;
    coords[(size_t)b * 2 * SAMP + SAMP + o] = yg;
  }
}

// ---------------------------------------------------------------------------
// Kernel 3: separable bilinear gather. One block per (b, c, oy); the two
// needed input rows are staged into LDS with coalesced 448-float loads, then
// the per-column gather hits LDS instead of strided global memory.
// ---------------------------------------------------------------------------
__global__ void masnet_sample_kernel(const float* __restrict__ data,
                                     const float* __restrict__ coords,
                                     float* __restrict__ out) {
  const int idx = blockIdx.x;       // (b*3 + c)*SAMP + oy
  const int oy = idx % SAMP;
  const int bc = idx / SAMP;
  const int b = bc / 3;

  __shared__ float srow0[INW];
  __shared__ float srow1[INW];

  const float ygv = coords[(size_t)b * 2 * SAMP + SAMP + oy];
  const float y = (ygv + 1.0f) * 0.5f * (float)(INW - 1);
  const float y0f = floorf(y);
  const float wy = y - y0f;
  const int y0 = min(max((int)y0f, 0), INW - 1);
  const int y1 = min(y0 + 1, INW - 1);

  const float* base = data + (size_t)bc * INW * INW;
  const float* r0 = base + (size_t)y0 * INW;
  const float* r1 = base + (size_t)y1 * INW;
  for (int j = threadIdx.x; j < INW; j += SAMP) {
    srow0[j] = r0[j];
    srow1[j] = r1[j];
  }
  __syncthreads();

  const int ox = threadIdx.x;
  const float xgv = coords[(size_t)b * 2 * SAMP + ox];
  const float x = (xgv + 1.0f) * 0.5f * (float)(INW - 1);
  const float x0f = floorf(x);
  const float wx = x - x0f;
  const int x0 = min(max((int)x0f, 0), INW - 1);
  const int x1 = min(x0 + 1, INW - 1);

  const float i00 = srow0[x0], i01 = srow0[x1];
  const float i10 = srow1[x0], i11 = srow1[x1];
  const float top = i00 * (1.0f - wx) + i01 * wx;
  const float bot = i10 * (1.0f - wx) + i11 * wx;
  out[(size_t)idx * SAMP + ox] = top * (1.0f - wy) + bot * wy;
}

// ---------------------------------------------------------------------------
extern "C" void kernel_launch(void* const* d_in, const int* in_sizes, int n_in,
                              void* d_out, int out_size, void* d_ws, size_t ws_size,
                              hipStream_t stream) {
  const float* data = (const float*)d_in[0];   // (64,3,448,448) f32
  const float* att  = (const float*)d_in[1];   // (64,448,448)   f32
  const float* fw   = (const float*)d_in[2];   // (447,)         f32
  float* outp = (float*)d_out;                 // (64,3,224,224) f32

  float* maps   = (float*)d_ws;                       // BATCH * 2 * 448
  float* coords = maps + (size_t)BATCH * 2 * INW;     // BATCH * 2 * 224

  masnet_reduce_kernel<<<BATCH, 256, 0, stream>>>(att, maps);
  masnet_grid_kernel<<<BATCH, 256, 0, stream>>>(maps, fw, coords);
  masnet_sample_kernel<<<BATCH * 3 * SAMP, SAMP, 0, stream>>>(data, coords, outp);
}